// WavetableSynth_24318104830782
// MI455X (gfx1250) — compile-verified
//
#include <hip/hip_runtime.h>
#include <hip/hip_bf16.h>
#include <stdint.h>

// ---------------- problem constants ----------------
#define N_WT      10
#define WT_LEN    512
#define FIR_TAPS  31
#define N_REFLECT 15
#define BB        16
#define LL        160000
#define CHUNK     4096
#define NCHUNKS   ((LL + CHUNK - 1) / CHUNK)   // 40
#define INC_SCALE 0.032f                        // WT_LEN / SR = 512/16000

typedef __attribute__((ext_vector_type(2))) float v2f;
typedef __attribute__((ext_vector_type(8))) float v8f;
typedef __attribute__((ext_vector_type(4))) int   v4i;

#define AS1 __attribute__((address_space(1)))
#define AS3 __attribute__((address_space(3)))

#if __has_builtin(__builtin_amdgcn_global_load_async_to_lds_b128) && \
    __has_builtin(__builtin_amdgcn_s_wait_asynccnt)
#define HAVE_ASYNC_LDS 1
#else
#define HAVE_ASYNC_LDS 0
#endif

// ============================================================================
// Kernel 1: FIR filter the wavetables on the WMMA path.
// out[w,i] = sum_t wt_pad[w, i+t] * h[t]  (circular pad by 15 each side)
// Blocked-Toeplitz matmul per 16-column tile:
//   D[w,n] = sum_{jr=0..47} wt_pad[w, i0+jr] * T[jr,n],  T[jr,n] = h[jr-n] (0..30)
// 12 chained V_WMMA_F32_16X16X4_F32 per wave. A zero-padded tap table s_hx
// makes every operand an unconditional LDS read (no EXEC juggling).
// Output stored transposed wtT[col*10 + w].
// ============================================================================
__global__ __launch_bounds__(1024) void fir_wmma_kernel(
    const float* __restrict__ wavetables,   // (10, 512)
    const float* __restrict__ fir_h,        // (31,)
    float* __restrict__ wtT)                // (512, 10) column-major result
{
    __shared__ float s_wt[N_WT * WT_LEN];
    __shared__ float s_hx[64];              // s_hx[j+15] = h[j] for j in [0,30], else 0
    const int tid = threadIdx.x;
    for (int i = tid; i < N_WT * WT_LEN; i += 1024) s_wt[i] = wavetables[i];
    if (tid < 64)
        s_hx[tid] = (tid >= N_REFLECT && tid < N_REFLECT + FIR_TAPS)
                        ? fir_h[tid - N_REFLECT] : 0.0f;
    __syncthreads();

    const int wave   = tid >> 5;        // 0..31 -> one 16-column tile each
    const int lane   = tid & 31;
    const int half   = lane >> 4;       // 0: lanes 0-15, 1: lanes 16-31
    const int lane16 = lane & 15;
    const int i0     = wave * 16;
    // Rows 10..15 of D are never stored, so A rows 10..15 may hold garbage;
    // just clamp the LDS index to stay in-bounds.
    const int mrow   = (lane16 < N_WT) ? lane16 : (lane16 - N_WT);

    v8f c = {};
#pragma unroll
    for (int kc = 0; kc < 12; ++kc) {
        // A 16x4 f32: lanes 0-15 hold K=0,1; lanes 16-31 hold K=2,3; M=lane16
        const int jbase = i0 + kc * 4 + 2 * half;            // absolute padded col
        v2f a;
        a.x = s_wt[mrow * WT_LEN + ((jbase + 0 + (WT_LEN - N_REFLECT)) & (WT_LEN - 1))];
        a.y = s_wt[mrow * WT_LEN + ((jbase + 1 + (WT_LEN - N_REFLECT)) & (WT_LEN - 1))];
        // B 4x16: VGPR j -> K=j (lanes 0-15) / K=j+2 (lanes 16-31); N=lane16
        const int kx = kc * 4 + 2 * half;                    // relative Toeplitz row
        v2f b;
        b.x = s_hx[kx - lane16 + N_REFLECT];
        b.y = s_hx[kx + 1 - lane16 + N_REFLECT];
        c = __builtin_amdgcn_wmma_f32_16x16x4_f32(false, a, false, b,
                                                  (short)0, c, false, false);
    }
    // D 16x16 f32: VGPR j -> M=j (lanes 0-15) / M=j+8 (lanes 16-31); N=lane16
    const int col = i0 + lane16;
#pragma unroll
    for (int j = 0; j < 8; ++j) {
        const int row = j + half * 8;
        if (row < N_WT) wtT[col * N_WT + row] = c[j];
    }
}

// ============================================================================
// Kernel 2: per-(batch, chunk) partial sums of the phase increments.
// ============================================================================
__global__ __launch_bounds__(256) void partial_kernel(
    const float* __restrict__ pitch, float* __restrict__ partials)
{
    const int cb = blockIdx.x, b = blockIdx.y, t = threadIdx.x;
    const long base   = (long)b * LL + (long)cb * CHUNK;
    const int  nvalid = (LL - cb * CHUNK < CHUNK) ? (LL - cb * CHUNK) : CHUNK;
    float s = 0.0f;
    for (int k = t; k < nvalid; k += 256) s += pitch[base + k] * INC_SCALE;
    __shared__ float red[256];
    red[t] = s;
    __syncthreads();
    for (int off = 128; off > 0; off >>= 1) {
        if (t < off) red[t] += red[t + off];
        __syncthreads();
    }
    if (t == 0) partials[b * NCHUNKS + cb] = red[0];
}

// ============================================================================
// Kernel 3: exclusive scan of chunk sums per batch row (tiny).
// ============================================================================
__global__ void scanbase_kernel(const float* __restrict__ partials,
                                float* __restrict__ bases)
{
    const int b = threadIdx.x;
    if (b < BB) {
        float run = 0.0f;
        for (int c = 0; c < NCHUNKS; ++c) {
            bases[b * NCHUNKS + c] = run;
            run += partials[b * NCHUNKS + c];
        }
    }
}

// ============================================================================
// One sample: phase wrap + table interpolation + 10-way attention dot.
// ============================================================================
__device__ __forceinline__ float sample_mix(float x,
                                            const float* __restrict__ a10,
                                            const float* s_wt)
{
    float r = fmodf(x, 512.0f);
    r = (r < 0.0f) ? r + 512.0f : r;
    r = (512.0f - r < 1e-5f) ? 0.0f : r;              // reference epsilon snap
    const float fl    = floorf(r);
    const float alpha = r - fl;
    const int ilo = (int)fl;
    const int ihi = ((int)ceilf(r)) & (WT_LEN - 1);
    const float2* a2  = (const float2*)a10;           // 8B-aligned (40B rows)
    const float2* lo2 = (const float2*)(s_wt + ilo * N_WT);
    const float2* hi2 = (const float2*)(s_wt + ihi * N_WT);
    float acc = 0.0f;
#pragma unroll
    for (int w = 0; w < 5; ++w) {
        const float2 at = a2[w], lo = lo2[w], hi = hi2[w];
        acc = fmaf(at.x, fmaf(alpha, hi.x - lo.x, lo.x), acc);
        acc = fmaf(at.y, fmaf(alpha, hi.y - lo.y, lo.y), acc);
    }
    return acc;
}

// ============================================================================
// Kernel 4: fused scan + interpolation + attention mix (bandwidth bound).
// ============================================================================
__global__ __launch_bounds__(256) void synth_kernel(
    const float* __restrict__ pitch,
    const float* __restrict__ envelope,
    const float* __restrict__ attention,
    const float* __restrict__ wtT,        // (512,10) filtered, column-major
    const float* __restrict__ bases,
    float* __restrict__ out)
{
    __shared__ float s_wt[WT_LEN * N_WT];     // 20 KB: [col*10 + w]
    __shared__ float s_scan[256];

    const int cb = blockIdx.x, b = blockIdx.y, t = threadIdx.x;

    // ---- stage filtered wavetables into LDS (async on gfx1250) ----
#if HAVE_ASYNC_LDS
    {
        // (global int4* src, shared int4* dst, imm offset, imm cpol)
        AS1 v4i* g = (AS1 v4i*)(uintptr_t)wtT;
        AS3 v4i* l = (AS3 v4i*)(uint32_t)(uintptr_t)&s_wt[0];
        for (int i = t; i < (WT_LEN * N_WT) / 4; i += 256) {
            __builtin_amdgcn_global_load_async_to_lds_b128(g + i, l + i, 0, 0);
        }
    }
#else
    for (int i = t; i < WT_LEN * N_WT; i += 256) s_wt[i] = wtT[i];
#endif

    // ---- blocked per-thread cumsum of increments (overlaps async copy) ----
    const int  base_l = cb * CHUNK + t * 16;
    const bool full   = (base_l + 16 <= LL);
    const float* prow = pitch + (long)b * LL;
    float inc[16], p0[16];
    float run = 0.0f;
    if (full) {
        const float4* pr4 = (const float4*)(prow + base_l);    // 64B aligned
        const float4* p04 = (const float4*)(pitch + base_l);   // batch-row 0
#pragma unroll
        for (int q = 0; q < 4; ++q) {
            const float4 pv = pr4[q];
            const float4 pz = p04[q];
            run += pv.x * INC_SCALE; inc[4*q+0] = run; p0[4*q+0] = pz.x * INC_SCALE;
            run += pv.y * INC_SCALE; inc[4*q+1] = run; p0[4*q+1] = pz.y * INC_SCALE;
            run += pv.z * INC_SCALE; inc[4*q+2] = run; p0[4*q+2] = pz.z * INC_SCALE;
            run += pv.w * INC_SCALE; inc[4*q+3] = run; p0[4*q+3] = pz.w * INC_SCALE;
        }
    } else {
#pragma unroll
        for (int k = 0; k < 16; ++k) {
            const int l = base_l + k;
            const float pv = (l < LL) ? prow[l]  : 0.0f;
            const float pz = (l < LL) ? pitch[l] : 0.0f;
            run += pv * INC_SCALE; inc[k] = run; p0[k] = pz * INC_SCALE;
        }
    }

    // ---- Hillis-Steele scan of the 256 thread totals (uniform barriers) ----
    s_scan[t] = run;
    __syncthreads();
    const float total = run;
    for (int off = 1; off < 256; off <<= 1) {
        const float v   = s_scan[t];
        const float add = (t >= off) ? s_scan[t - off] : 0.0f;
        __syncthreads();
        s_scan[t] = v + add;
        __syncthreads();
    }
    const float excl  = s_scan[t] - total;
    const float Sbase = bases[b * NCHUNKS + cb] + excl;

#if HAVE_ASYNC_LDS
    __builtin_amdgcn_s_wait_asynccnt(0);
#endif
    __syncthreads();                       // s_wt fully resident for all waves

    // ---- stream attention/envelope, interpolate, mix ----
    const float* erow = envelope + (long)b * LL;
    const float* arow = attention + (long)b * LL * N_WT;
    float* orow = out + (long)b * LL;

    if (base_l < LL)
        __builtin_prefetch(arow + (long)base_l * N_WT, 0, 0);

    if (full) {
        float res[16];
#pragma unroll
        for (int k = 0; k < 16; ++k) {
            const long l = base_l + k;
            // index = cumsum(b,l) - increment(0,l)   (faithful torch quirk)
            res[k] = sample_mix(Sbase + inc[k] - p0[k], arow + l * N_WT, s_wt);
        }
#pragma unroll
        for (int q = 0; q < 4; ++q) {
            const float4 ev = *(const float4*)(erow + base_l + 4 * q);
            const float4 ov = make_float4(res[4*q+0] * ev.x, res[4*q+1] * ev.y,
                                          res[4*q+2] * ev.z, res[4*q+3] * ev.w);
            *(float4*)(orow + base_l + 4 * q) = ov;
        }
    } else {
#pragma unroll
        for (int k = 0; k < 16; ++k) {
            const int l = base_l + k;
            if (l < LL) {
                const float acc = sample_mix(Sbase + inc[k] - p0[k],
                                             arow + (long)l * N_WT, s_wt);
                orow[l] = acc * erow[l];
            }
        }
    }
}

// ============================================================================
extern "C" void kernel_launch(void* const* d_in, const int* in_sizes, int n_in,
                              void* d_out, int out_size, void* d_ws, size_t ws_size,
                              hipStream_t stream)
{
    const float* pitch      = (const float*)d_in[0];   // (16,160000,1)
    const float* envelope   = (const float*)d_in[1];   // (16,160000,1)
    const float* attention  = (const float*)d_in[2];   // (16,160000,10)
    const float* wavetables = (const float*)d_in[3];   // (10,512)
    const float* fir_h      = (const float*)d_in[4];   // (31,)
    float* out = (float*)d_out;

    // scratch layout (floats): wtT[5120] | partials[16*40] | bases[16*40]
    float* wtT      = (float*)d_ws;
    float* partials = wtT + N_WT * WT_LEN;
    float* bases    = partials + BB * NCHUNKS;

    fir_wmma_kernel<<<1, 1024, 0, stream>>>(wavetables, fir_h, wtT);
    partial_kernel<<<dim3(NCHUNKS, BB), 256, 0, stream>>>(pitch, partials);
    scanbase_kernel<<<1, 32, 0, stream>>>(partials, bases);
    synth_kernel<<<dim3(NCHUNKS, BB), 256, 0, stream>>>(
        pitch, envelope, attention, wtT, bases, out);
}